// VolIntegratedLSTM_67619965108298
// MI455X (gfx1250) — compile-verified
//
#include <hip/hip_runtime.h>
#include <math.h>

typedef float v2f __attribute__((ext_vector_type(2)));
typedef float v8f __attribute__((ext_vector_type(8)));

#define NB_RED 512
#define RED_THREADS 256

// ---------- fast hardware transcendentals ----------
__device__ __forceinline__ float fexp(float x)   { return __builtin_amdgcn_exp2f(x * 1.4426950408889634f); }
__device__ __forceinline__ float fln(float x)    { return __builtin_amdgcn_logf(x) * 0.6931471805599453f; }
__device__ __forceinline__ float frsq(float x)   { return __builtin_amdgcn_rsqf(x); }
__device__ __forceinline__ float fsqrtf_(float x){ return __builtin_amdgcn_sqrtf(x); }
// native v_tanh_f32 on gfx1250 (TRANS op)
__device__ __forceinline__ float ftanh_(float x) { return __builtin_amdgcn_tanhf(x); }
__device__ __forceinline__ float sigmoidf_(float x) { return fmaf(0.5f, ftanh_(0.5f * x), 0.5f); }
__device__ __forceinline__ float softplusf_(float x){ return fln(1.f + fexp(x)); }  // arg in (-1,1): safe

__device__ __forceinline__ float rdlane(float v, int l) {
  return __int_as_float(__builtin_amdgcn_readlane(__float_as_int(v), l));
}

// ---------- stage 1: deterministic partial reduction for var(y) ----------
__global__ void vils_reduce_kernel(const float* __restrict__ y, int T, float* __restrict__ ws) {
  __shared__ float s_sum[RED_THREADS];
  __shared__ float s_sq[RED_THREADS];
  int tid = threadIdx.x;
  long long i = (long long)blockIdx.x * RED_THREADS + tid;
  long long stride = (long long)gridDim.x * RED_THREADS;
  float sum = 0.f, sq = 0.f;
  for (; i < T; i += stride) { float v = y[i]; sum += v; sq = fmaf(v, v, sq); }
  s_sum[tid] = sum; s_sq[tid] = sq;
  __syncthreads();
  for (int off = RED_THREADS / 2; off > 0; off >>= 1) {
    if (tid < off) { s_sum[tid] += s_sum[tid + off]; s_sq[tid] += s_sq[tid + off]; }
    __syncthreads();
  }
  if (tid == 0) { ws[blockIdx.x] = s_sum[0]; ws[NB_RED + blockIdx.x] = s_sq[0]; }
}

// uniform parameters (compiler keeps these in SGPRs; gfx1250 even has s_add_f32 etc.)
struct P13 {
  float Ai, Af, Ao, Ac;   // combined eta coefficient: W_g[0] + U_g
  float Bi, Bf, Bo, Bc;   // z coefficient: W_g[3]
  float om, al, ph, la, ga;
};

// one recurrence step; bases + yt arrive as uniform (readlane/SGPR) operands
__device__ __forceinline__ void vils_step(float& eta, float& cst, float& hv, float& z,
                                          float bi_, float bf_, float bo_, float bc_,
                                          float yt, const P13& p,
                                          float& zn, float& hn) {
  const float pre_i = fmaf(p.Ai, eta, fmaf(p.Bi, z, bi_));
  const float pre_f = fmaf(p.Af, eta, fmaf(p.Bf, z, bf_));
  const float pre_o = fmaf(p.Ao, eta, fmaf(p.Bo, z, bo_));
  const float pre_c = fmaf(p.Ac, eta, fmaf(p.Bc, z, bc_));
  const float it = sigmoidf_(pre_i);
  const float ft = sigmoidf_(pre_f);
  const float ot = sigmoidf_(pre_o);
  const float gt = ftanh_(pre_c);
  const float ct = fmaf(ft, cst, it * gt);
  const float etan = softplusf_(ot * ftanh_(ct));
  const float sh = fsqrtf_(hv);                       // h-chain runs parallel to LSTM chain
  hn = fmaf(p.ph, hv - p.om, p.om) + p.al * (fmaf(z, z, -1.f) - 2.f * p.ga * sh * z);
  const float irs = frsq(etan);
  zn = fmaf(-p.la, etan, yt) * irs;
  eta = etan; cst = ct; hv = hn; z = zn;
}

__device__ __forceinline__ v8f base_wmma(v2f a, v2f b) {
  v8f c = {0.f, 0.f, 0.f, 0.f, 0.f, 0.f, 0.f, 0.f};
  return __builtin_amdgcn_wmma_f32_16x16x4_f32(false, a, false, b, (short)0, c, false, false);
}

// ---------- stage 2: single-wave serial scan, software-pipelined 16-step tiles ----------
__global__ void __launch_bounds__(32, 1)
vils_scan_kernel(const float* __restrict__ y, const float* __restrict__ rv,
                 const float* __restrict__ Wi, const float* __restrict__ Ui, const float* __restrict__ bi,
                 const float* __restrict__ Wf, const float* __restrict__ Uf, const float* __restrict__ bf,
                 const float* __restrict__ Wo, const float* __restrict__ Uo, const float* __restrict__ bo,
                 const float* __restrict__ Wc, const float* __restrict__ Uc, const float* __restrict__ bc,
                 const float* __restrict__ omega_p, const float* __restrict__ alpha_p,
                 const float* __restrict__ phi_p, const float* __restrict__ lam_p,
                 const float* __restrict__ gam_p,
                 int T, const float* __restrict__ ws,
                 float* __restrict__ z_out, float* __restrict__ h_out) {
  const int lane = threadIdx.x;
  const int gl = lane & 15;

  // ---- finish the variance reduction (fixed order -> deterministic) ----
  float sum = 0.f, sq = 0.f;
  for (int i = lane; i < NB_RED; i += 32) { sum += ws[i]; sq += ws[NB_RED + i]; }
  for (int off = 16; off > 0; off >>= 1) {
    sum += __shfl_xor(sum, off, 32);
    sq  += __shfl_xor(sq,  off, 32);
  }
  const float invT = 1.f / (float)T;
  const float mean = sum * invT;
  const float var  = sq * invT - mean * mean;

  // ---- uniform parameters ----
  P13 p;
  p.Ai = Wi[0] + Ui[0]; p.Bi = Wi[3];
  p.Af = Wf[0] + Uf[0]; p.Bf = Wf[3];
  p.Ao = Wo[0] + Uo[0]; p.Bo = Wo[3];
  p.Ac = Wc[0] + Uc[0]; p.Bc = Wc[3];
  p.om = omega_p[0]; p.al = alpha_p[0]; p.ph = phi_p[0]; p.la = lam_p[0]; p.ga = gam_p[0];

  // ---- initial state ----
  float eta = var, cst = 0.f, hv = var;
  float z = fmaf(-p.la, eta, y[0]) * frsq(eta);
  if (lane == 0) { z_out[0] = z; h_out[0] = var; }

  // ---- loop-invariant WMMA A-matrix: rows = gates, A[g][*] = [W_g[1], W_g[2], b_g, 0] ----
  // f32 16x4 A layout: lanes 0-15 hold K=0 (v0) / K=1 (v1); lanes 16-31 hold K=2 (v0) / K=3 (v1).
  float w1 = 0.f, w2 = 0.f, bbv = 0.f;
  if      (gl == 0) { w1 = Wi[1]; w2 = Wi[2]; bbv = bi[0]; }
  else if (gl == 1) { w1 = Wf[1]; w2 = Wf[2]; bbv = bf[0]; }
  else if (gl == 2) { w1 = Wo[1]; w2 = Wo[2]; bbv = bo[0]; }
  else if (gl == 3) { w1 = Wc[1]; w2 = Wc[2]; bbv = bc[0]; }
  v2f amat;
  amat.x = (lane < 16) ? w1 : bbv;
  amat.y = (lane < 16) ? w2 : 0.f;

  const int nsteps = T - 1;
  const int last = T - 1;

  // ---- prologue: tile 0 loads + WMMA ----
  // yv[lane] = y[t0+lane]: lanes 0..15 are y_prev, lane k+1 supplies y_t for step k.
  int i0 = lane; if (i0 > last) i0 = last;
  float yv  = y[i0];
  float rvv = rv[gl];                 // gl <= 15 < T: always in range for tile 0
  v2f bm0; bm0.x = (lane < 16) ? yv : 1.f; bm0.y = (lane < 16) ? rvv : 0.f;
  v8f acc = base_wmma(amat, bm0);

  int t0 = 0;
  for (; t0 + 16 <= nsteps; t0 += 16) {
    // -- issue next tile's loads first (latency hidden under this tile's chain) --
    const int tn = t0 + 16;
    int iy = tn + lane; if (iy > last) iy = last;
    int ir = tn + gl;   if (ir > last) ir = last;
    const float yv_n  = y[iy];
    const float rvv_n = rv[ir];
    if (tn + 64 < nsteps) {
      // locality hint 3 -> WGP-scope prefetch: pulls lines into the near cache too
      __builtin_prefetch(y  + tn + 64 + lane, 0, 3);
      __builtin_prefetch(rv + tn + 64 + lane, 0, 3);
    }

    // -- fully-unrolled 16-step chain on current tile (readlane broadcasts, off-chain) --
    float z_keep = 0.f, h_keep = 0.f;
    #pragma unroll
    for (int k = 0; k < 16; ++k) {
      const float bi_ = rdlane(acc[0], k);
      const float bf_ = rdlane(acc[1], k);
      const float bo_ = rdlane(acc[2], k);
      const float bc_ = rdlane(acc[3], k);
      const float yt  = rdlane(yv, k + 1);
      float zn, hn;
      vils_step(eta, cst, hv, z, bi_, bf_, bo_, bc_, yt, p, zn, hn);
      z_keep = (lane == k) ? zn : z_keep;
      h_keep = (lane == k) ? hn : h_keep;
    }
    if (lane < 16) {
      z_out[t0 + lane + 1] = z_keep;
      h_out[t0 + lane + 1] = h_keep;
    }

    // -- next tile's WMMA at the bottom: its load-wait is hidden by the chain above --
    v2f bm; bm.x = (lane < 16) ? yv_n : 1.f; bm.y = (lane < 16) ? rvv_n : 0.f;
    acc = base_wmma(amat, bm);
    yv = yv_n; rvv = rvv_n;
  }

  // ---- tail tile (< 16 steps), executes at most once ----
  if (t0 < nsteps) {
    const int r = nsteps - t0;      // 1..15; acc/yv already hold this tile (clamped loads are real y values)
    float z_keep = 0.f, h_keep = 0.f;
    for (int k = 0; k < r; ++k) {   // uniform dynamic readlane index: legal for v_readlane
      const float bi_ = rdlane(acc[0], k);
      const float bf_ = rdlane(acc[1], k);
      const float bo_ = rdlane(acc[2], k);
      const float bc_ = rdlane(acc[3], k);
      const float yt  = rdlane(yv, k + 1);
      float zn, hn;
      vils_step(eta, cst, hv, z, bi_, bf_, bo_, bc_, yt, p, zn, hn);
      z_keep = (lane == k) ? zn : z_keep;
      h_keep = (lane == k) ? hn : h_keep;
    }
    if (lane < r) {
      z_out[t0 + lane + 1] = z_keep;
      h_out[t0 + lane + 1] = h_keep;
    }
  }
}

extern "C" void kernel_launch(void* const* d_in, const int* in_sizes, int n_in,
                              void* d_out, int out_size, void* d_ws, size_t ws_size,
                              hipStream_t stream) {
  const float* y   = (const float*)d_in[0];
  const float* rv  = (const float*)d_in[1];
  const float* Wi  = (const float*)d_in[2];
  const float* Ui  = (const float*)d_in[3];
  const float* bi  = (const float*)d_in[4];
  const float* Wf  = (const float*)d_in[5];
  const float* Uf  = (const float*)d_in[6];
  const float* bf  = (const float*)d_in[7];
  const float* Wo  = (const float*)d_in[8];
  const float* Uo  = (const float*)d_in[9];
  const float* bo  = (const float*)d_in[10];
  const float* Wc  = (const float*)d_in[11];
  const float* Uc  = (const float*)d_in[12];
  const float* bc  = (const float*)d_in[13];
  const float* omg = (const float*)d_in[14];
  const float* alp = (const float*)d_in[15];
  const float* phi = (const float*)d_in[16];
  const float* lam = (const float*)d_in[17];
  const float* gam = (const float*)d_in[18];
  const int T = in_sizes[0];
  float* ws = (float*)d_ws;
  float* zo = (float*)d_out;
  float* ho = zo + T;

  vils_reduce_kernel<<<NB_RED, RED_THREADS, 0, stream>>>(y, T, ws);
  vils_scan_kernel<<<1, 32, 0, stream>>>(y, rv, Wi, Ui, bi, Wf, Uf, bf, Wo, Uo, bo,
                                         Wc, Uc, bc, omg, alp, phi, lam, gam,
                                         T, ws, zo, ho);
}